// RFFTLayer2d_67405216743944
// MI455X (gfx1250) — compile-verified
//
#include <hip/hip_runtime.h>
#include <hip/hip_bf16.h>

// RFFT over 4x4 blocks == D = T(16x16 cos matrix) * X(16 blocks) per tile.
// Memory-bound (419 MB @ 23.3 TB/s ~ 18us). WMMA_F32_16X16X4_F32 x4 per tile
// does the contraction in the matrix pipe; the cosine A-matrix is built once
// per wave (cospif is ~150 VALU ops) and amortized over 16 tiles so the VALU
// cost stays far below the HBM roofline.

typedef __attribute__((ext_vector_type(2))) float v2f;
typedef __attribute__((ext_vector_type(8))) float v8f;

#define WAVES_PER_BLOCK 8

__global__ __launch_bounds__(256)
void rfft16_block_wmma_kernel(const float* __restrict__ x, float* __restrict__ out) {
  const int lane = threadIdx.x & 31;
  const int wave = threadIdx.x >> 5;
  const int t    = blockIdx.x * WAVES_PER_BLOCK + wave;  // (batch, block-row) id

  const int H = t & 255;          // block row (0..255)
  const int b = t >> 8;           // batch (0..63)

  const int n    = lane & 15;     // matrix column / A-matrix row (M)
  const int half = lane >> 4;     // lane half selects K/M sub-ranges

  // ---- A operand: T[m][c] = cos(2*pi*m*c/16) = cospi(m*c/8), fp32 ----
  // 16x4 f32 A layout: VGPR0 = K{0|2}, VGPR1 = K{1|3} selected by lane half;
  // K-chunk j covers c = 4j + 2*half + v.  Built ONCE per wave, reused 16x.
  const float mf = (float)n;
  const float c0 = (float)(2 * half);
  v2f a0, a1, a2, a3;
  a0.x = cospif(mf * (c0 +  0.0f) * 0.125f);
  a0.y = cospif(mf * (c0 +  1.0f) * 0.125f);
  a1.x = cospif(mf * (c0 +  4.0f) * 0.125f);
  a1.y = cospif(mf * (c0 +  5.0f) * 0.125f);
  a2.x = cospif(mf * (c0 +  8.0f) * 0.125f);
  a2.y = cospif(mf * (c0 +  9.0f) * 0.125f);
  a3.x = cospif(mf * (c0 + 12.0f) * 0.125f);
  a3.y = cospif(mf * (c0 + 13.0f) * 0.125f);

  // ---- B operand: K-chunk j == spatial row j of the 4x4 block ----
  // Lane (n,half) loads the contiguous pair x[b, 4H+j, 64*wt + 4n + 2*half + {0,1}]
  // -> one b64 NT load per lane per chunk; the wave exactly tiles each 256B
  // row segment, fully coalesced.
  const float* p = x + ((size_t)b * 1024 + (size_t)(4 * H)) * 1024
                     + (size_t)(4 * n + 2 * half);
  float* o = out + (((size_t)b * 256 + (size_t)H) * 256 + (size_t)n) * 9;

#pragma unroll 4
  for (int wt = 0; wt < 16; ++wt) {
    v2f b0 = __builtin_nontemporal_load((const v2f*)(p + 0 * 1024));
    v2f b1 = __builtin_nontemporal_load((const v2f*)(p + 1 * 1024));
    v2f b2 = __builtin_nontemporal_load((const v2f*)(p + 2 * 1024));
    v2f b3 = __builtin_nontemporal_load((const v2f*)(p + 3 * 1024));

    v8f acc;
#pragma unroll
    for (int i = 0; i < 8; ++i) acc[i] = 0.0f;   // folds into inline-0 SRC2

    // D = sum_j A_j(16x4) * B_j(4x16), fp32 accumulate (RNE), EXEC all-ones.
    acc = __builtin_amdgcn_wmma_f32_16x16x4_f32(false, a0, false, b0, (short)0, acc, false, false);
    acc = __builtin_amdgcn_wmma_f32_16x16x4_f32(false, a1, false, b1, (short)0, acc, false, false);
    acc = __builtin_amdgcn_wmma_f32_16x16x4_f32(false, a2, false, b2, (short)0, acc, false, false);
    acc = __builtin_amdgcn_wmma_f32_16x16x4_f32(false, a3, false, b3, (short)0, acc, false, false);

    // ---- Store: lane holds column n, VGPR v = row m = 8*half + v; keep m<=8.
    if (half == 0) {
#pragma unroll
      for (int v = 0; v < 8; ++v)
        __builtin_nontemporal_store(acc[v], o + v);     // m = 0..7
    } else {
      __builtin_nontemporal_store(acc[0], o + 8);       // m = 8
    }

    p += 64;        // next 16-block group along W (256 B)
    o += 16 * 9;    // 16 blocks * 9 freqs
  }
}

extern "C" void kernel_launch(void* const* d_in, const int* in_sizes, int n_in,
                              void* d_out, int out_size, void* d_ws, size_t ws_size,
                              hipStream_t stream) {
  const float* x   = (const float*)d_in[0];
  float*       out = (float*)d_out;

  // 64 batches * 256 block-rows = 16384 waves, each looping over 16 tiles.
  const int total_rows = 64 * 256;
  dim3 grid(total_rows / WAVES_PER_BLOCK);   // 2048
  dim3 block(256);
  rfft16_block_wmma_kernel<<<grid, block, 0, stream>>>(x, out);
}